// GIN_78005196030467
// MI455X (gfx1250) — compile-verified
//
#include <hip/hip_runtime.h>
#include <stdint.h>

typedef float v2f __attribute__((ext_vector_type(2)));
typedef float v8f __attribute__((ext_vector_type(8)));

#define GIN_N_NODES 100000
#define GIN_N_EDGES 1600000

// ======================= CSR construction =======================
__global__ void gin_zero_ints(int* __restrict__ p, int n) {
  int i = blockIdx.x * blockDim.x + threadIdx.x;
  if (i < n) p[i] = 0;
}

__global__ void gin_hist(const int* __restrict__ dstIdx, int* __restrict__ deg) {
  int e = blockIdx.x * blockDim.x + threadIdx.x;
  if (e < GIN_N_EDGES) atomicAdd(&deg[dstIdx[e]], 1);
}

// single-workgroup chunked exclusive scan over n bins (n = 100000)
__global__ void gin_scan(const int* __restrict__ deg, int* __restrict__ rowstart,
                         int* __restrict__ cursor, int n) {
  __shared__ int buf[1024];
  __shared__ int carry;
  if (threadIdx.x == 0) carry = 0;
  __syncthreads();
  for (int base = 0; base < n; base += 1024) {
    int i = base + (int)threadIdx.x;
    int v = (i < n) ? deg[i] : 0;
    buf[threadIdx.x] = v;
    __syncthreads();
#pragma unroll
    for (int off = 1; off < 1024; off <<= 1) {
      int t = (threadIdx.x >= (unsigned)off) ? buf[threadIdx.x - off] : 0;
      __syncthreads();
      buf[threadIdx.x] += t;
      __syncthreads();
    }
    int excl = buf[threadIdx.x] - v + carry;
    if (i < n) { rowstart[i] = excl; cursor[i] = excl; }
    __syncthreads();
    if (threadIdx.x == 1023) carry += buf[1023];
    __syncthreads();
  }
  if (threadIdx.x == 0) rowstart[n] = carry;   // == N_EDGES
}

__global__ void gin_build_csr(const int* __restrict__ srcIdx, const int* __restrict__ dstIdx,
                              int* __restrict__ cursor, int* __restrict__ esrc) {
  int e = blockIdx.x * blockDim.x + threadIdx.x;
  if (e >= GIN_N_EDGES) return;
  int p = atomicAdd(&cursor[dstIdx[e]], 1);
  esrc[p] = srcIdx[e];
}

// ============ gather: agg[n] = (1+eps)*feat[n] + sum_{e in CSR[n]} feat[esrc[e]] ============
// One wave32 per destination node; register accumulation, zero float atomics.
template <int CH>
__global__ void gin_gather(const float* __restrict__ feat, const int* __restrict__ rowstart,
                           const int* __restrict__ esrc, const float* __restrict__ eps,
                           float* __restrict__ agg) {
  int node = (blockIdx.x * blockDim.x + threadIdx.x) >> 5;
  int lane = threadIdx.x & 31;
  if (node >= GIN_N_NODES) return;
  constexpr int V = CH / 128;          // float4 chunks per lane
  int beg = rowstart[node], end = rowstart[node + 1];
  float s = 1.0f + eps[0];
  const float4* self = (const float4*)(feat + (size_t)node * CH);
  float4 acc[V];
#pragma unroll
  for (int v = 0; v < V; ++v) {
    float4 t = self[lane + 32 * v];
    acc[v].x = t.x * s; acc[v].y = t.y * s; acc[v].z = t.z * s; acc[v].w = t.w * s;
  }
  for (int e = beg; e < end; ++e) {
    int srcn = esrc[e];
    const float4* fp = (const float4*)(feat + (size_t)srcn * CH);
#pragma unroll
    for (int v = 0; v < V; ++v) {
      float4 t = fp[lane + 32 * v];
      acc[v].x += t.x; acc[v].y += t.y; acc[v].z += t.z; acc[v].w += t.w;
    }
  }
  float4* op = (float4*)(agg + (size_t)node * CH);
#pragma unroll
  for (int v = 0; v < V; ++v) op[lane + 32 * v] = acc[v];
}

// ======================= weight repack for b64 B-loads =======================
// Wp[k2][col] = { W[2*k2][col], W[2*k2+1][col] }
__global__ void gin_repack_w(const float* __restrict__ W, float2* __restrict__ Wp,
                             int K, int NCOLS) {
  int i = blockIdx.x * blockDim.x + threadIdx.x;
  int total = (K / 2) * NCOLS;
  if (i >= total) return;
  int k2 = i / NCOLS, col = i - k2 * NCOLS;
  Wp[i] = make_float2(W[(size_t)(2 * k2) * NCOLS + col],
                      W[(size_t)(2 * k2 + 1) * NCOLS + col]);
}

// ======================= GEMM: out = [relu](A @ W + bias) =======================
// One wave32 per block; tile = 32 rows x 64 cols via V_WMMA_F32_16X16X4_F32
// (2x4 = 8 v8f accumulators = 64 VGPRs; no spills).
// Layouts per CDNA5 ISA 7.12.2:
//   A 16x4 : row = lane&15; lanes 0-15 hold K={k,k+1}, lanes 16-31 hold K={k+2,k+3}
//   B 4x16 : col = lane&15; same K split across lane halves (repacked -> one b64 load)
//   C 16x16: col = lane&15; vgpr r -> row r (lanes 0-15) / row r+8 (lanes 16-31)
template <int K, int NCOLS, bool RELU>
__global__ __launch_bounds__(32, 1)
void gin_gemm(const float* __restrict__ A, const float2* __restrict__ Wp,
              const float* __restrict__ bias, float* __restrict__ out) {
  const int lane = threadIdx.x;
  const int l    = lane & 15;
  const int half = lane >> 4;
  const int m0   = blockIdx.x * 32;  // two 16-row tiles
  const int n0   = blockIdx.y * 64;  // four 16-col tiles

  const float* a0 = A + (size_t)(m0 + l)      * K + 2 * half;
  const float* a1 = A + (size_t)(m0 + 16 + l) * K + 2 * half;

  v8f acc[2][4];
#pragma unroll
  for (int mt = 0; mt < 2; ++mt)
#pragma unroll
    for (int t = 0; t < 4; ++t)
      acc[mt][t] = (v8f){0.f, 0.f, 0.f, 0.f, 0.f, 0.f, 0.f, 0.f};

#pragma unroll 4
  for (int k = 0; k < K; k += 4) {
    v2f av0 = *(const v2f*)(a0 + k);
    v2f av1 = *(const v2f*)(a1 + k);
    const float2* wp = Wp + (size_t)(k / 2 + half) * NCOLS + n0 + l;
#pragma unroll
    for (int t = 0; t < 4; ++t) {
      v2f bv = *(const v2f*)(wp + t * 16);
      acc[0][t] = __builtin_amdgcn_wmma_f32_16x16x4_f32(
          false, av0, false, bv, (short)0, acc[0][t], false, false);
      acc[1][t] = __builtin_amdgcn_wmma_f32_16x16x4_f32(
          false, av1, false, bv, (short)0, acc[1][t], false, false);
    }
  }

  // epilogue: bias (+ relu), write 16x16 tiles
#pragma unroll
  for (int mt = 0; mt < 2; ++mt) {
#pragma unroll
    for (int t = 0; t < 4; ++t) {
      int col  = n0 + t * 16 + l;
      float bv = bias[col];
      float* op = out + (size_t)(m0 + mt * 16 + 8 * half) * NCOLS + col;
#pragma unroll
      for (int r = 0; r < 8; ++r) {
        float v = acc[mt][t][r] + bv;
        if (RELU) v = fmaxf(v, 0.0f);
        op[(size_t)r * NCOLS] = v;
      }
    }
  }
}

extern "C" void kernel_launch(void* const* d_in, const int* in_sizes, int n_in,
                              void* d_out, int out_size, void* d_ws, size_t ws_size,
                              hipStream_t stream) {
  const float* x    = (const float*)d_in[0];
  const int*   ei   = (const int*)d_in[1];
  const float* eps1 = (const float*)d_in[2];
  const float* W1a  = (const float*)d_in[3];
  const float* b1a  = (const float*)d_in[4];
  const float* W1b  = (const float*)d_in[5];
  const float* b1b  = (const float*)d_in[6];
  const float* eps2 = (const float*)d_in[7];
  const float* W2a  = (const float*)d_in[8];
  const float* b2a  = (const float*)d_in[9];
  const float* W2b  = (const float*)d_in[10];
  const float* b2b  = (const float*)d_in[11];
  float* out = (float*)d_out;

  const int* src = ei;               // edge_index[0]
  const int* dst = ei + GIN_N_EDGES; // edge_index[1]

  // ---- workspace layout ----
  const size_t NB = (size_t)GIN_N_NODES * 256;
  float* Wagg = (float*)d_ws;
  float* Wmid = Wagg + NB;
  float* Whid = Wmid + NB;
  int* deg      = (int*)(Whid + NB);
  int* rowstart = deg + GIN_N_NODES;           // N+1 entries
  int* cursor   = rowstart + GIN_N_NODES + 1;
  int* esrc     = cursor + GIN_N_NODES;
  float2* wp1a  = (float2*)(((uintptr_t)(esrc + GIN_N_EDGES) + 15) & ~(uintptr_t)15);
  float2* wp1b  = wp1a + 64 * 256;   // K/2 * NCOLS
  float2* wp2a  = wp1b + 128 * 256;
  float2* wp2b  = wp2a + 128 * 256;  // 128*128

  const int eb  = (GIN_N_EDGES + 255) / 256;
  const int nwb = (GIN_N_NODES * 32 + 255) / 256;

  // ---- build CSR (once, reused by both layers) ----
  gin_zero_ints<<<(GIN_N_NODES + 255) / 256, 256, 0, stream>>>(deg, GIN_N_NODES);
  gin_hist<<<eb, 256, 0, stream>>>(dst, deg);
  gin_scan<<<1, 1024, 0, stream>>>(deg, rowstart, cursor, GIN_N_NODES);
  gin_build_csr<<<eb, 256, 0, stream>>>(src, dst, cursor, esrc);

  // ---- repack weights for b64 B-operand loads ----
  gin_repack_w<<<(64 * 256 + 255) / 256, 256, 0, stream>>>(W1a, wp1a, 128, 256);
  gin_repack_w<<<(128 * 256 + 255) / 256, 256, 0, stream>>>(W1b, wp1b, 256, 256);
  gin_repack_w<<<(128 * 256 + 255) / 256, 256, 0, stream>>>(W2a, wp2a, 256, 256);
  gin_repack_w<<<(128 * 128 + 255) / 256, 256, 0, stream>>>(W2b, wp2b, 256, 128);

  // ---- layer 1 ----
  gin_gather<128><<<nwb, 256, 0, stream>>>(x, rowstart, esrc, eps1, Wagg);
  gin_gemm<128, 256, true>
      <<<dim3(GIN_N_NODES / 32, 256 / 64), 32, 0, stream>>>(Wagg, wp1a, b1a, Wmid);
  gin_gemm<256, 256, true>   // outer relu fused: relu(h@W1b + b1b)
      <<<dim3(GIN_N_NODES / 32, 256 / 64), 32, 0, stream>>>(Wmid, wp1b, b1b, Whid);

  // ---- layer 2 ----
  gin_gather<256><<<nwb, 256, 0, stream>>>(Whid, rowstart, esrc, eps2, Wagg);
  gin_gemm<256, 256, true>
      <<<dim3(GIN_N_NODES / 32, 256 / 64), 32, 0, stream>>>(Wagg, wp2a, b2a, Wmid);
  gin_gemm<256, 128, false>
      <<<dim3(GIN_N_NODES / 32, 128 / 64), 32, 0, stream>>>(Wmid, wp2b, b2b, out);
}